// RollingHashPyramid_87016037416983
// MI455X (gfx1250) — compile-verified
//
#include <hip/hip_runtime.h>

typedef __attribute__((ext_vector_type(2))) float v2f;
typedef __attribute__((ext_vector_type(4))) float v4f;
typedef __attribute__((ext_vector_type(8))) float v8f;

// Hash primes (reference BASE / EXT32); all arithmetic valid in u32 since keys are mod 2^16.
__device__ constexpr unsigned BASEC[16] = {
    2654435761u, 2246822519u, 3266489917u, 2028178513u, 1220703125u,
    1610612741u, 805306457u,  402653189u,  3674653429u, 2860486313u,
    1073676287u, 2971215073u, 1500450271u, 3267000013u, 2654435789u, 4049292737u};
// EXT32 primes pre-XORed with 0xDEADBEEF (scale-5 prime list).
__device__ constexpr unsigned P5C[32] = {
    2654435761u ^ 0xDEADBEEFu, 2246822519u ^ 0xDEADBEEFu, 3266489917u ^ 0xDEADBEEFu,
    2028178513u ^ 0xDEADBEEFu, 1220703125u ^ 0xDEADBEEFu, 1610612741u ^ 0xDEADBEEFu,
    805306457u  ^ 0xDEADBEEFu, 402653189u  ^ 0xDEADBEEFu, 3674653429u ^ 0xDEADBEEFu,
    2860486313u ^ 0xDEADBEEFu, 1073676287u ^ 0xDEADBEEFu, 2971215073u ^ 0xDEADBEEFu,
    1500450271u ^ 0xDEADBEEFu, 3267000013u ^ 0xDEADBEEFu, 2654435789u ^ 0xDEADBEEFu,
    4049292737u ^ 0xDEADBEEFu, 2246822531u ^ 0xDEADBEEFu, 3266489927u ^ 0xDEADBEEFu,
    2028178519u ^ 0xDEADBEEFu, 1220703133u ^ 0xDEADBEEFu, 1610612759u ^ 0xDEADBEEFu,
    805306463u  ^ 0xDEADBEEFu, 402653201u  ^ 0xDEADBEEFu, 3674653441u ^ 0xDEADBEEFu,
    2860486319u ^ 0xDEADBEEFu, 1073676311u ^ 0xDEADBEEFu, 2971215091u ^ 0xDEADBEEFu,
    1500450277u ^ 0xDEADBEEFu, 3267000023u ^ 0xDEADBEEFu, 2654435801u ^ 0xDEADBEEFu,
    4049292751u ^ 0xDEADBEEFu, 2246822537u ^ 0xDEADBEEFu};

#define T_DIM 4096
#define PPB 128        // positions per block
#define STRIDE 68      // LDS row stride in floats (68 mod 64 = 4 -> conflict-free b64 reads)

__global__ __launch_bounds__(128, 4)
void rhp_kernel(const int* __restrict__ tokens,
                const float* __restrict__ tables,
                const float* __restrict__ Wc,
                float* __restrict__ out,
                int nIter)   // ==16 at runtime; opaque -> gather loops stay rolled (unroll 4)
{
    __shared__ unsigned tk[256];            // tokens[t0-128 .. t0+127]
    __shared__ unsigned keys_lds[8][PPB];
    __shared__ unsigned ck_lds[PPB];        // cond_key per position
    __shared__ unsigned cklut[256];         // sign-bit byte -> XOR of COND_PRIMES
    __shared__ __align__(16) float short_lds[PPB * STRIDE];
    __shared__ __align__(16) float wc_lds[16 * STRIDE];   // Wc rows 0..7; rows 8..15 = zero pad

    const int tid = threadIdx.x;
    const int bid = blockIdx.x;
    const int b   = bid >> 5;               // T/PPB = 32 blocks per row
    const int t0  = (bid & 31) << 7;
    const long rowBase = (long)b * T_DIM;

    // ---- Phase 1: tokens -> LDS; cond-prime LUT; Wc -> LDS (zero-padded to 16 rows)
    for (int j = tid; j < 256; j += 128) {
        int g = t0 - 128 + j;
        tk[j] = (g >= 0) ? (unsigned)tokens[rowBase + g] : 0u;
        unsigned v = 0;
        #pragma unroll
        for (int i = 0; i < 8; ++i)
            if (j & (1 << i)) v ^= BASEC[i];   // constexpr -> pure VALU, no mem loads
        cklut[j] = v;
    }
    for (int idx = tid; idx < 16 * STRIDE; idx += 128) {
        int row = idx / STRIDE;
        int col = idx - row * STRIDE;
        wc_lds[idx] = (row < 8 && col < 64) ? Wc[row * 64 + col] : 0.0f;
    }
    __syncthreads();

    // ---- Phase 2: 8 hash keys per position, all in u32 (exact mod 2^16)
    {
        const int base = 128 + tid;
        unsigned h = 0;
        #pragma unroll
        for (int i = 0; i < 16; ++i) {      // scales 0..4 are XOR-prefixes of BASE chain
            h ^= tk[base - 1 - i] * BASEC[i];
            if (i == 0)  keys_lds[0][tid] = h & 0xFFFFu;
            if (i == 1)  keys_lds[1][tid] = h & 0xFFFFu;
            if (i == 3)  keys_lds[2][tid] = h & 0xFFFFu;
            if (i == 7)  keys_lds[3][tid] = h & 0xFFFFu;
            if (i == 15) keys_lds[4][tid] = h & 0xFFFFu;
        }
        unsigned h5 = 0, h6 = 0, h7 = 0;
        #pragma unroll 8
        for (int i = 0; i < 32; ++i)  h5 ^= tk[base - 1 - i] * P5C[i];
        keys_lds[5][tid] = h5 & 0xFFFFu;
        #pragma unroll 8
        for (int i = 0; i < 64; ++i)  h6 ^= tk[base - 1 - i] * (BASEC[i & 7] ^ 0xCAFEBABEu);
        keys_lds[6][tid] = h6 & 0xFFFFu;
        #pragma unroll 8
        for (int i = 0; i < 128; ++i) h7 ^= tk[base - 1 - i] * (BASEC[i & 7] ^ 0x8BADF00Du);
        keys_lds[7][tid] = h7 & 0xFFFFu;
    }
    __syncthreads();

    const v4f* tab4 = reinterpret_cast<const v4f*>(tables);
    v4f* out4 = reinterpret_cast<v4f*>(out);
    const long outPosBase = (rowBase + t0) * 8;   // in 16-float rows

    // ---- Phase 3: short gather (scales 0..3) -> LDS tile + direct NT output
    #pragma unroll 4
    for (int j = 0; j < nIter; ++j) {
        int flat   = j * 128 + tid;       // 0..2047 = 512 rows x 4 float4 pieces
        int rowIdx = flat >> 2;
        int piece  = flat & 3;
        int scale  = rowIdx >> 7;         // 0..3
        int pos    = rowIdx & 127;
        unsigned key = keys_lds[scale][pos];
        v4f v = tab4[((size_t)(scale << 16) + key) * 4 + piece];
        *reinterpret_cast<v4f*>(&short_lds[pos * STRIDE + scale * 16 + piece * 4]) = v;
        __builtin_nontemporal_store(v, &out4[(outPosBase + (long)pos * 8 + scale) * 4 + piece]);
    }
    __syncthreads();

    // ---- Phase 4: cond logits via V_WMMA_F32_16X16X4_F32 (exact f32 -> safe sign bits)
    {
        const int lane = tid & 31;
        const int wave = tid >> 5;
        const int n  = lane & 15;         // N column (logit index); also M row for A operand
        const int hi = lane >> 4;         // half-wave: K pair select
        const float* bp = &wc_lds[n * STRIDE + 2 * hi];   // B[kk][n]=Wc[n][4k+kk]; rows>=8 zero
        #pragma unroll
        for (int tt = 0; tt < 2; ++tt) {  // each wave owns its own 32 positions = 2 tiles
            const int mb = wave * 32 + tt * 16;
            const float* ap = &short_lds[(mb + n) * STRIDE + 2 * hi];
            v8f c = {0.f, 0.f, 0.f, 0.f, 0.f, 0.f, 0.f, 0.f};
            #pragma unroll
            for (int k = 0; k < 16; ++k) {
                v2f a  = *reinterpret_cast<const v2f*>(ap + 4 * k);
                v2f bb = *reinterpret_cast<const v2f*>(bp + 4 * k);
                c = __builtin_amdgcn_wmma_f32_16x16x4_f32(
                        false, a, false, bb, (short)0, c, false, false);
            }
            // D layout: vgpr v @ half hi -> position mb + v + 8*hi, column n.
            // Ballot collects sign bits of column n across lanes: bits[7:0] = pos mb+v,
            // bits[23:16] = pos mb+v+8. One lane per position writes cond_key via LUT.
            #pragma unroll
            for (int v = 0; v < 8; ++v) {
#if __has_builtin(__builtin_amdgcn_ballot_w32)
                unsigned mask = __builtin_amdgcn_ballot_w32(c[v] > 0.0f);
                unsigned bits = (mask >> (16 * hi)) & 0xFFu;
                if (n == v)
                    ck_lds[mb + v + 8 * hi] = cklut[bits];
#else
                if (n < 8 && c[v] > 0.0f)
                    atomicXor(&ck_lds[mb + v + 8 * hi], BASEC[n]);
#endif
            }
        }
    }
    __syncthreads();

    // ---- Phase 5: long gather (scales 4..7) at key ^ cond_key -> NT output
    #pragma unroll 4
    for (int j = 0; j < nIter; ++j) {
        int flat   = j * 128 + tid;
        int rowIdx = flat >> 2;
        int piece  = flat & 3;
        int scale  = 4 + (rowIdx >> 7);   // 4..7
        int pos    = rowIdx & 127;
        unsigned key = (keys_lds[scale][pos] ^ ck_lds[pos]) & 0xFFFFu;
        v4f v = tab4[((size_t)(scale << 16) + key) * 4 + piece];
        __builtin_nontemporal_store(v, &out4[(outPosBase + (long)pos * 8 + scale) * 4 + piece]);
    }
}

extern "C" void kernel_launch(void* const* d_in, const int* in_sizes, int n_in,
                              void* d_out, int out_size, void* d_ws, size_t ws_size,
                              hipStream_t stream) {
    (void)in_sizes; (void)n_in; (void)out_size; (void)d_ws; (void)ws_size;
    const int*   tokens = (const int*)d_in[0];
    const float* tables = (const float*)d_in[1];
    const float* Wc     = (const float*)d_in[2];
    float*       out    = (float*)d_out;
    const int B = 32, T = 4096;
    dim3 grid(B * (T / PPB)), block(128);
    hipLaunchKernelGGL(rhp_kernel, grid, block, 0, stream, tokens, tables, Wc, out, 16);
}